// GATForecast_84043920048138
// MI455X (gfx1250) — compile-verified
//
#include <hip/hip_runtime.h>
#include <hip/hip_bf16.h>
#include <math.h>

// ---------------------------------------------------------------------------
// GAT forecast pipeline for gfx1250 (MI455X), fp32 end-to-end.
// Dense GEMMs via V_WMMA_F32_16X16X4_F32 (wave32, 16-row tiles per wave).
// Edge softmax/aggregation via coalesced atomics (L2-resident, the real cost).
// ---------------------------------------------------------------------------

typedef float v2f __attribute__((ext_vector_type(2)));
typedef float v8f __attribute__((ext_vector_type(8)));

#define DEVINL __device__ __forceinline__

// ---- order-preserving float<->uint mapping for atomicMax on f32 ------------
DEVINL unsigned gat_fmapu(float f) {
  unsigned b = __float_as_uint(f);
  return b ^ ((b & 0x80000000u) ? 0xFFFFFFFFu : 0x80000000u);
}
DEVINL float gat_fumap(unsigned u) {
  unsigned b = (u & 0x80000000u) ? (u ^ 0x80000000u) : ~u;
  return __uint_as_float(b);
}
DEVINL float gat_lrelu(float x) { return x > 0.f ? x : 0.2f * x; }

// ---- WMMA f32 GEMM: C[M,NC] = act(A[M,KD] @ B[KD,NC] + bias) ---------------
// One wave computes rows [16*wave, 16*wave+16) x all NC columns.
// A frag (16x4 f32): lane<16 -> M=lane, K={k0,k0+1}; lane>=16 -> K={k0+2,k0+3}
// B frag (4x16 f32): lane half selects K pair, lane&15 selects N column.
// D (16x16 f32, 8 VGPRs): M = r + 8*(lane>>4), N = lane&15.
template <int KD, int NC, int ACT>
__global__ void gat_wmma_mm(const float* __restrict__ A,
                            const float* __restrict__ B,
                            const float* __restrict__ bias,
                            float* __restrict__ C, int M) {
  const int wave = blockIdx.x * (blockDim.x >> 5) + (threadIdx.x >> 5);
  const int lane = threadIdx.x & 31;
  const int tiles = M >> 4;
  if (wave >= tiles) return;                 // wave-uniform: EXEC stays all-1s
  const int row0 = wave << 4;
  const int g = lane >> 4;
  const int l16 = lane & 15;
  constexpr int NT = NC / 16;

  v8f acc[NT];
#pragma unroll
  for (int t = 0; t < NT; ++t)
#pragma unroll
    for (int r = 0; r < 8; ++r) acc[t][r] = 0.f;

  const float* arow = A + (size_t)(row0 + l16) * KD + 2 * g;
  for (int k0 = 0; k0 < KD; k0 += 4) {
    v2f a;
    a.x = arow[k0];
    a.y = arow[k0 + 1];
    const float* brow = B + (size_t)(k0 + 2 * g) * NC + l16;
#pragma unroll
    for (int t = 0; t < NT; ++t) {
      v2f b;
      b.x = brow[t * 16];
      b.y = brow[NC + t * 16];
      acc[t] = __builtin_amdgcn_wmma_f32_16x16x4_f32(
          false, a, false, b, (short)0, acc[t], false, false);
    }
  }

#pragma unroll
  for (int t = 0; t < NT; ++t) {
    const int col = t * 16 + l16;
    const float bb = bias ? bias[col] : 0.f;
#pragma unroll
    for (int r = 0; r < 8; ++r) {
      float v = acc[t][r] + bb;
      if (ACT == 1) v = fmaxf(v, 0.f);
      C[(size_t)(row0 + r + 8 * g) * NC + col] = v;
    }
  }
}

// ---- per-node attention coefficients: a_s[n,h] = xh[n,h,:].att_s[h,:] ------
__global__ void gat_attn_dots(const float* __restrict__ xh,
                              const float* __restrict__ att_s,
                              const float* __restrict__ att_d,
                              float* __restrict__ asn, float* __restrict__ adn,
                              int n, int heads, int ch) {
  int t = blockIdx.x * blockDim.x + threadIdx.x;
  if (t >= n * heads) return;
  int h = t % heads;
  int node = t / heads;
  const float* v = xh + (size_t)node * heads * ch + h * ch;
  const float* ws = att_s + h * ch;
  const float* wd = att_d + h * ch;
  float s = 0.f, d = 0.f;
  for (int c = 0; c < ch; ++c) {
    float xv = v[c];
    s = fmaf(xv, ws[c], s);
    d = fmaf(xv, wd[c], d);
  }
  asn[t] = s;
  adn[t] = d;
}

// ---- edge pass 1: segment max over dst --------------------------------------
__global__ void gat_edge_max(const long long* __restrict__ ei, int Eraw, int Etot,
                             const float* __restrict__ asn,
                             const float* __restrict__ adn,
                             unsigned* __restrict__ maxu, int heads) {
  int e = blockIdx.x * blockDim.x + threadIdx.x;
  if (e >= Etot) return;
  int s, d;
  if (e < Eraw) { s = (int)ei[e]; d = (int)ei[Eraw + e]; }
  else          { s = e - Eraw; d = s; }
  for (int h = 0; h < heads; ++h) {
    float ev = gat_lrelu(asn[s * heads + h] + adn[d * heads + h]);
    atomicMax(&maxu[d * heads + h], gat_fmapu(ev));
  }
}

// ---- edge pass 2: exp(e - max), segment sum; stash unnormalized weight -----
__global__ void gat_edge_expsum(const long long* __restrict__ ei, int Eraw, int Etot,
                                const float* __restrict__ asn,
                                const float* __restrict__ adn,
                                const unsigned* __restrict__ maxu,
                                float* __restrict__ sum,
                                float* __restrict__ alpha, int heads) {
  int e = blockIdx.x * blockDim.x + threadIdx.x;
  if (e >= Etot) return;
  int s, d;
  if (e < Eraw) { s = (int)ei[e]; d = (int)ei[Eraw + e]; }
  else          { s = e - Eraw; d = s; }
  for (int h = 0; h < heads; ++h) {
    float ev = gat_lrelu(asn[s * heads + h] + adn[d * heads + h]);
    float w = expf(ev - gat_fumap(maxu[d * heads + h]));
    alpha[e * heads + h] = w;
    atomicAdd(&sum[d * heads + h], w);
  }
}

// ---- edge pass 3: out[dst] += alpha * xh[src]  (coalesced per-edge strips) -
template <int HE, int CH>
__global__ void gat_edge_msg(const long long* __restrict__ ei, int Eraw, int total,
                             const float* __restrict__ xh,
                             const float* __restrict__ alpha,
                             const float* __restrict__ sum,
                             float* __restrict__ out) {
  constexpr int F = HE * CH;
  int t = blockIdx.x * blockDim.x + threadIdx.x;
  if (t >= total) return;
  int e = t / F;
  int j = t - e * F;                // channel index; a wave covers one edge strip
  int h = j / CH;
  int s, d;
  if (e < Eraw) { s = (int)ei[e]; d = (int)ei[Eraw + e]; }
  else          { s = e - Eraw; d = s; }
  float a = alpha[e * HE + h] / (sum[d * HE + h] + 1e-16f);
  atomicAdd(&out[(size_t)d * F + j], xh[(size_t)s * F + j] * a);
}

// ---- bias + ELU (in place) --------------------------------------------------
__global__ void gat_bias_elu(float* __restrict__ io, const float* __restrict__ bias,
                             int total, int mask) {
  int t = blockIdx.x * blockDim.x + threadIdx.x;
  if (t >= total) return;
  float v = io[t] + bias[t & mask];
  io[t] = v > 0.f ? v : (expf(v) - 1.f);
}

// ---- global mean pool -------------------------------------------------------
__global__ void gat_pool(const float* __restrict__ h,
                         const long long* __restrict__ batch,
                         float* __restrict__ pooled, float* __restrict__ cnts,
                         int total) {
  int t = blockIdx.x * blockDim.x + threadIdx.x;
  if (t >= total) return;
  int node = t >> 6;                 // GH = 64
  int j = t & 63;
  int g = (int)batch[node];
  atomicAdd(&pooled[g * 64 + j], h[t]);
  if (j == 0) atomicAdd(&cnts[g], 1.f);
}

__global__ void gat_pool_div(const float* __restrict__ pooled,
                             const float* __restrict__ cnts,
                             float* __restrict__ hg, int total) {
  int t = blockIdx.x * blockDim.x + threadIdx.x;
  if (t >= total) return;
  hg[t] = pooled[t] / fmaxf(cnts[t >> 6], 1.f);
}

// ---- three output heads -----------------------------------------------------
__global__ void gat_heads(const float* __restrict__ z,
                          const float* Wc, const float* bc,
                          const float* Wr, const float* br,
                          const float* Wv, const float* bv,
                          float* __restrict__ out, int G) {
  int g = blockIdx.x * blockDim.x + threadIdx.x;
  if (g >= G) return;
  const float* zr = z + (size_t)g * 64;
  float c = bc[0], r = br[0], v = bv[0];
  for (int i = 0; i < 64; ++i) {
    float zz = zr[i];
    c = fmaf(zz, Wc[i], c);
    r = fmaf(zz, Wr[i], r);
    v = fmaf(zz, Wv[i], v);
  }
  out[g] = c;
  out[G + g] = r;
  out[2 * G + g] = fmaxf(v, 0.f) + log1pf(expf(-fabsf(v)));  // stable softplus
}

// ---- fills ------------------------------------------------------------------
__global__ void gat_fill_f32(float* p, float v, int n) {
  int t = blockIdx.x * blockDim.x + threadIdx.x;
  if (t < n) p[t] = v;
}
__global__ void gat_fill_u32(unsigned* p, unsigned v, int n) {
  int t = blockIdx.x * blockDim.x + threadIdx.x;
  if (t < n) p[t] = v;
}

// ---------------------------------------------------------------------------
extern "C" void kernel_launch(void* const* d_in, const int* in_sizes, int n_in,
                              void* d_out, int out_size, void* d_ws, size_t ws_size,
                              hipStream_t stream) {
  const float*     x     = (const float*)d_in[0];
  const long long* ei    = (const long long*)d_in[1];
  const long long* batch = (const long long*)d_in[2];
  const float* Wp  = (const float*)d_in[3];
  const float* bp  = (const float*)d_in[4];
  const float* W1  = (const float*)d_in[5];
  const float* as1 = (const float*)d_in[6];
  const float* ad1 = (const float*)d_in[7];
  const float* b1  = (const float*)d_in[8];
  const float* W2  = (const float*)d_in[9];
  const float* as2 = (const float*)d_in[10];
  const float* ad2 = (const float*)d_in[11];
  const float* b2  = (const float*)d_in[12];
  const float* Wm1 = (const float*)d_in[13];
  const float* bm1 = (const float*)d_in[14];
  const float* Wm2 = (const float*)d_in[15];
  const float* bm2 = (const float*)d_in[16];
  const float* Wc  = (const float*)d_in[17];
  const float* bc  = (const float*)d_in[18];
  const float* Wr  = (const float*)d_in[19];
  const float* br  = (const float*)d_in[20];
  const float* Wv  = (const float*)d_in[21];
  const float* bv  = (const float*)d_in[22];
  float* out = (float*)d_out;

  const int N  = in_sizes[0] / 128;   // 50000 (divisible by 16)
  const int E  = in_sizes[1] / 2;     // 800000
  const int E2 = E + N;               // edges + self loops
  const int G  = out_size / 3;        // 1024

  // workspace layout (aliased across pipeline stages)
  float* w = (float*)d_ws;
  float* bufA = w;                 w += (size_t)N * 64;   // h0, later gat2 out (h2)
  float* bufB = w;                 w += (size_t)N * 128;  // xh1, later xh2
  float* asn  = w;                 w += (size_t)N * 4;
  float* adn  = w;                 w += (size_t)N * 4;
  unsigned* maxu = (unsigned*)w;   w += (size_t)N * 4;
  float* sum  = w;                 w += (size_t)N * 4;
  float* alpha = w;                w += (size_t)E2 * 4;
  float* bufH = w;                 w += (size_t)N * 128;  // gat1 out (h1)
  float* pooled = w;               w += (size_t)G * 64;
  float* cnts = w;                 w += (size_t)G;
  float* hg   = w;                 w += (size_t)G * 64;
  float* z1   = w;                 w += (size_t)G * 64;
  float* z2   = w;                 w += (size_t)G * 64;

  const int TB = 256;
  const int WPB = TB / 32;
  auto nb = [](long long t, int b) { return (int)((t + b - 1) / b); };
  auto mmB = [&](int M) { return ((M >> 4) + WPB - 1) / WPB; };
  const unsigned NEG_INF_MAPPED = 0x007FFFFFu;  // fmapu(-inf)

  // ---- stage 1: node projection ----
  gat_wmma_mm<128, 64, 1><<<mmB(N), TB, 0, stream>>>(x, Wp, bp, bufA, N);

  // ---- stage 2: GAT layer 1 (4 heads x 32 ch, concat) ----
  gat_wmma_mm<64, 128, 0><<<mmB(N), TB, 0, stream>>>(bufA, W1, nullptr, bufB, N);
  gat_attn_dots<<<nb((long long)N * 4, TB), TB, 0, stream>>>(bufB, as1, ad1, asn, adn, N, 4, 32);
  gat_fill_u32<<<nb(N * 4, TB), TB, 0, stream>>>(maxu, NEG_INF_MAPPED, N * 4);
  gat_fill_f32<<<nb(N * 4, TB), TB, 0, stream>>>(sum, 0.f, N * 4);
  gat_fill_f32<<<nb((long long)N * 128, TB), TB, 0, stream>>>(bufH, 0.f, N * 128);
  gat_edge_max<<<nb(E2, TB), TB, 0, stream>>>(ei, E, E2, asn, adn, maxu, 4);
  gat_edge_expsum<<<nb(E2, TB), TB, 0, stream>>>(ei, E, E2, asn, adn, maxu, sum, alpha, 4);
  gat_edge_msg<4, 32><<<nb((long long)E2 * 128, TB), TB, 0, stream>>>(ei, E, E2 * 128, bufB, alpha, sum, bufH);
  gat_bias_elu<<<nb((long long)N * 128, TB), TB, 0, stream>>>(bufH, b1, N * 128, 127);

  // ---- stage 3: GAT layer 2 (1 head x 64 ch) ----
  gat_wmma_mm<128, 64, 0><<<mmB(N), TB, 0, stream>>>(bufH, W2, nullptr, bufB, N);
  gat_attn_dots<<<nb(N, TB), TB, 0, stream>>>(bufB, as2, ad2, asn, adn, N, 1, 64);
  gat_fill_u32<<<nb(N, TB), TB, 0, stream>>>(maxu, NEG_INF_MAPPED, N);
  gat_fill_f32<<<nb(N, TB), TB, 0, stream>>>(sum, 0.f, N);
  gat_fill_f32<<<nb((long long)N * 64, TB), TB, 0, stream>>>(bufA, 0.f, N * 64);
  gat_edge_max<<<nb(E2, TB), TB, 0, stream>>>(ei, E, E2, asn, adn, maxu, 1);
  gat_edge_expsum<<<nb(E2, TB), TB, 0, stream>>>(ei, E, E2, asn, adn, maxu, sum, alpha, 1);
  gat_edge_msg<1, 64><<<nb((long long)E2 * 64, TB), TB, 0, stream>>>(ei, E, E2 * 64, bufB, alpha, sum, bufA);
  gat_bias_elu<<<nb((long long)N * 64, TB), TB, 0, stream>>>(bufA, b2, N * 64, 63);

  // ---- stage 4: global mean pool ----
  gat_fill_f32<<<nb(G * 64, TB), TB, 0, stream>>>(pooled, 0.f, G * 64);
  gat_fill_f32<<<nb(G, TB), TB, 0, stream>>>(cnts, 0.f, G);
  gat_pool<<<nb((long long)N * 64, TB), TB, 0, stream>>>(bufA, batch, pooled, cnts, N * 64);
  gat_pool_div<<<nb(G * 64, TB), TB, 0, stream>>>(pooled, cnts, hg, G * 64);

  // ---- stage 5: MLP + heads ----
  gat_wmma_mm<64, 64, 1><<<mmB(G), TB, 0, stream>>>(hg, Wm1, bm1, z1, G);
  gat_wmma_mm<64, 64, 1><<<mmB(G), TB, 0, stream>>>(z1, Wm2, bm2, z2, G);
  gat_heads<<<nb(G, TB), TB, 0, stream>>>(z2, Wc, bc, Wr, br, Wv, bv, out, G);
}